// Multixcorr_depthwise_44727789420780
// MI455X (gfx1250) — compile-verified
//
#include <hip/hip_runtime.h>
#include <hip/hip_bf16.h>

// CDNA5 / gfx1250 depthwise cross-correlation via V_WMMA_F32_16X16X4_F32,
// with async global->LDS staging (GLOBAL_LOAD_ASYNC_TO_LDS_B128 / ASYNCcnt).
//
// Per (b,c) plane (32x32 x, 8x8 k, 25x25 out), per input row r:
//   A(j,q) = x(r, j+q)   (M = output col j, K = tap q)
//   B(q,p) = k(p,q)      (N = kernel row p; cols p>=8 zero)
//   C(j,p) = sum_q x(r,j+q) k(p,q)  --> accumulates into out(r-p, j)
// Two M-tiles (j 0..15, 16..31) x two chained K-steps (q 0..3, 4..7).

typedef __attribute__((ext_vector_type(2))) float v2f;
typedef __attribute__((ext_vector_type(4))) float v4f;
typedef __attribute__((ext_vector_type(8))) float v8f;
typedef int v4i_vs __attribute__((vector_size(4 * sizeof(int))));

#define NB 128
#define NC 256
#define NPLANES (NB * NC)      // 32768
#define IH 32
#define IW 32
#define KH 8
#define KW 8
#define OH 25
#define OW 25
#define OPLANE (OH * OW)       // 625

#define WAVES_PER_BLOCK 4
#define LDS_ROW 40             // 32 cols + 8 zero pad (so j+q up to 38 is safe)
#define XLDS_SIZE (IH * LDS_ROW)       // 1280 floats
#define OACC_W 32
#define OACC_SIZE (OH * OACC_W)        // 800 floats
#define PLANE_LDS (XLDS_SIZE + OACC_SIZE)  // 2080 floats per wave

// Async DMA of 16B global -> LDS, bypassing VGPRs. Tracked with ASYNCcnt.
__device__ static inline void async_ld_b128(const float* gsrc, float* ldst) {
    __builtin_amdgcn_global_load_async_to_lds_b128(
        (__attribute__((address_space(1))) v4i_vs*)(uintptr_t)gsrc,
        (__attribute__((address_space(3))) v4i_vs*)(unsigned)(uintptr_t)ldst,
        /*offset=*/0, /*cpol=*/0);
}

__device__ static inline void wait_async0() {
    asm volatile("s_wait_asynccnt 0x0" ::: "memory");
}

__global__ __launch_bounds__(32 * WAVES_PER_BLOCK)
void dwxcorr_wmma_f32(const float* __restrict__ x,
                      const float* __restrict__ ker,
                      const float* __restrict__ w,
                      float* __restrict__ out)
{
    __shared__ float lds[WAVES_PER_BLOCK * PLANE_LDS];

    const int lane    = threadIdx.x & 31;
    const int wave    = threadIdx.x >> 5;
    const int plane   = blockIdx.x * WAVES_PER_BLOCK + wave;   // wave-uniform
    if (plane >= NPLANES) return;                              // uniform per wave

    float* xs   = &lds[wave * PLANE_LDS];
    float* oacc = xs + XLDS_SIZE;

    const float* xp = x   + (size_t)plane * (IH * IW);
    const float* kp = ker + (size_t)plane * 64;

    const int half = lane >> 4;        // 0: K={0,1}, 1: K={2,3} within a WMMA
    const int nm   = lane & 15;        // M (=j_local) for A, N (=p) for B

    // ---- Stage x plane into LDS via async DMA (lane = row, 8 chunks/lane
    // over the wave = 8 x 512B). No VGPR round-trip; overlaps with zeroing.
    {
        float* row = xs + lane * LDS_ROW;
        #pragma unroll
        for (int c = 0; c < IW / 4; ++c)
            async_ld_b128(xp + lane * IW + c * 4, row + c * 4);
        // zero the 8 pad columns (plain DS stores)
        #pragma unroll
        for (int c = IW; c < LDS_ROW; ++c) row[c] = 0.0f;
    }

    // ---- Zero the output accumulator (DScnt path, overlaps with async) ----
    for (int i = lane; i < OACC_SIZE; i += 32) oacc[i] = 0.0f;

    // ---- Load kernel into B-matrix layout (two K-steps) ----
    // B element (K = 2*half + v, N = nm) = k(p=nm, q = 4*s + 2*half + v)
    v2f Bk0 = {0.0f, 0.0f};
    v2f Bk1 = {0.0f, 0.0f};
    if (nm < KH) {
        const float* kr = kp + nm * KW;
        Bk0.x = kr[0 + 2 * half]; Bk0.y = kr[1 + 2 * half];
        Bk1.x = kr[4 + 2 * half]; Bk1.y = kr[5 + 2 * half];
    }

    // Async staging must land before the A-matrix DS reads below.
    wait_async0();

    // ---- Main loop over input rows ----
    const int  p      = nm;                  // kernel row this lane's C column holds
    float* const dstb = oacc + 8 * half;     // col base j_local = v + 8*half

    for (int r = 0; r < IH; ++r) {
        const float* xrow = xs + r * LDS_ROW;

        const int b0 = 0  + nm + 2 * half;   // tile 0 A base (j + K-offset)
        const int b1 = 16 + nm + 2 * half;   // tile 1 A base

        v2f a00; a00.x = xrow[b0];     a00.y = xrow[b0 + 1];
        v2f a01; a01.x = xrow[b0 + 4]; a01.y = xrow[b0 + 5];
        v2f a10; a10.x = xrow[b1];     a10.y = xrow[b1 + 1];
        v2f a11; a11.x = xrow[b1 + 4]; a11.y = xrow[b1 + 5];

        v8f Cz = {0.f, 0.f, 0.f, 0.f, 0.f, 0.f, 0.f, 0.f};

        // tile 0: chained K-steps (q 0..3 then 4..7)
        v8f C0 = __builtin_amdgcn_wmma_f32_16x16x4_f32(
            false, a00, false, Bk0, (short)0, Cz, false, false);
        C0 = __builtin_amdgcn_wmma_f32_16x16x4_f32(
            false, a01, false, Bk1, (short)0, C0, false, false);

        // tile 1
        v8f C1 = __builtin_amdgcn_wmma_f32_16x16x4_f32(
            false, a10, false, Bk0, (short)0, Cz, false, false);
        C1 = __builtin_amdgcn_wmma_f32_16x16x4_f32(
            false, a11, false, Bk1, (short)0, C1, false, false);

        // Accumulate C(j, p) into out(r - p, j). Each valid lane owns 8
        // consecutive cols of one output row per tile -> no lane collisions;
        // same-wave DS ordering serializes across r iterations.
        const int orow = r - p;
        if (p < KH && orow >= 0 && orow < OH) {
            v4f* d = (v4f*)(dstb + orow * OACC_W);      // 32B-aligned
            v4f c0lo = __builtin_shufflevector(C0, C0, 0, 1, 2, 3);
            v4f c0hi = __builtin_shufflevector(C0, C0, 4, 5, 6, 7);
            v4f c1lo = __builtin_shufflevector(C1, C1, 0, 1, 2, 3);
            v4f c1hi = __builtin_shufflevector(C1, C1, 4, 5, 6, 7);
            d[0] = d[0] + c0lo;          // tile0 cols j_local+0..3
            d[1] = d[1] + c0hi;          // tile0 cols j_local+4..7
            d[4] = d[4] + c1lo;          // tile1 cols 16+j_local+0..3
            d[5] = d[5] + c1hi;          // tile1 cols 16+j_local+4..7
        }
    }

    // ---- Scale by sum(softmax(weight)) exactly as the reference computes ----
    float w0 = w[0], w1 = w[1], w2 = w[2];
    float mx = fmaxf(w0, fmaxf(w1, w2));
    float e0 = __expf(w0 - mx), e1 = __expf(w1 - mx), e2 = __expf(w2 - mx);
    float s  = e0 + e1 + e2;
    float scale = e0 / s + e1 / s + e2 / s;

    // ---- Store 25x25 outputs ----
    float* op = out + (size_t)plane * OPLANE;
    for (int i = lane; i < OPLANE; i += 32) {
        int orow = i / OW;
        int ocol = i - orow * OW;
        op[i] = scale * oacc[orow * OACC_W + ocol];
    }
}

extern "C" void kernel_launch(void* const* d_in, const int* in_sizes, int n_in,
                              void* d_out, int out_size, void* d_ws, size_t ws_size,
                              hipStream_t stream) {
    (void)in_sizes; (void)n_in; (void)out_size; (void)d_ws; (void)ws_size;
    const float* x   = (const float*)d_in[0];
    const float* ker = (const float*)d_in[1];
    const float* w   = (const float*)d_in[2];
    float* out       = (float*)d_out;

    dim3 block(32 * WAVES_PER_BLOCK);
    dim3 grid(NPLANES / WAVES_PER_BLOCK);   // 8192 blocks, 1 wave = 1 plane
    hipLaunchKernelGGL(dwxcorr_wmma_f32, grid, block, 0, stream, x, ker, w, out);
}